// HarmonicSynth_64055142252776
// MI455X (gfx1250) — compile-verified
//
#include <hip/hip_runtime.h>
#include <hip/hip_bf16.h>

typedef float v2f __attribute__((ext_vector_type(2)));
typedef float v8f __attribute__((ext_vector_type(8)));

#define BATCH   16
#define TKNOT   250
#define NHARM   100
#define NSAMP   48000

// ---------------------------------------------------------------------------
// Kernel 1 (setup, tiny): per-batch knot prefix sums of f0_up and the
// per-(batch,harmonic) frequency-stretch coefficients.
//   K[b][s] = sum_{t < 192*s + 96} f0_up[b][t]
//     (exact: K[0]=96*f[0]; K[s+1]=K[s]+96*(f[s]+f[s+1]); K[249]=S[47903])
//   C[b][k-1] = k * sqrt(1 + inh[b]*k^2)   (f64, hoisted out of hot loop)
// ---------------------------------------------------------------------------
__global__ void hsynth_setup_kernel(const float* __restrict__ f0,
                                    const float* __restrict__ inh,
                                    double* __restrict__ K,
                                    double* __restrict__ C) {
    const int tid = threadIdx.x;

    if (tid < BATCH) {
        const int b = tid;
        const float* f = f0 + b * TKNOT;
        double* Kb = K + b * TKNOT;
        double acc = 96.0 * (double)f[0];
        Kb[0] = acc;
        for (int s = 0; s < TKNOT - 1; ++s) {
            acc += 96.0 * ((double)f[s] + (double)f[s + 1]);
            Kb[s + 1] = acc;
        }
    }

    // 16*100 = 1600 stretch coefficients, strided over the 256 threads
    for (int i = tid; i < BATCH * NHARM; i += blockDim.x) {
        const int b = i / NHARM;
        const int k = (i % NHARM) + 1;
        const double kd = (double)k;
        C[i] = kd * sqrt(1.0 + (double)inh[b] * kd * kd);
    }
}

// ---------------------------------------------------------------------------
// Kernel 2: synthesis. One wave32 per (batch, 16-sample tile).
// Lane (L&15) owns sample t0+(L&15); halves own harmonic sub-pairs.
// Reduction over harmonics done on the WMMA pipe:
//   C += ones(16x4) * B(4x16),  B[kk, t] = a*mask*sin(2*pi * frac(c_k*PhiT_t))
// Every row of C ends up holding the full 100-harmonic sum for its column.
// Phase is tracked in TURNS (v_sin_f32 consumes turns): PhiT = S/SR.
// ---------------------------------------------------------------------------
__global__ void __launch_bounds__(256)
hsynth_synth_kernel(const float* __restrict__ ha,   // (B, T, N)
                    const float* __restrict__ f0,   // (B, T)
                    const double* __restrict__ K,   // (B, T) knot prefix
                    const double* __restrict__ C,   // (B, N) stretch coeffs
                    float* __restrict__ out) {      // (B, 1, NSAMP)
    const int gtid = blockIdx.x * blockDim.x + threadIdx.x;
    const int wave = gtid >> 5;
    const int lane = threadIdx.x & 31;
    const int half = lane >> 4;

    const int b    = wave / (NSAMP / 16);
    const int tile = wave % (NSAMP / 16);
    const int t    = tile * 16 + (lane & 15);

    const float*  f0b = f0 + b * TKNOT;
    const double* Kb  = K + b * TKNOT;
    const double* Cb  = C + b * NHARM;

    // --- PhiT[b,t] and interpolation coefficients (closed form, no scan) ---
    // Tile boundaries (multiples of 16) never straddle the regions
    // [0,95], [96,47903], [47904,47999], so this branch is wave-uniform.
    int i0, i1;
    float w;
    double S;
    if (t <= 95) {                       // pos clipped to 0
        i0 = 0; i1 = 0; w = 0.0f;
        S = (double)(t + 1) * (double)f0b[0];
    } else if (t >= 47904) {             // pos clipped to T-1
        i0 = TKNOT - 1; i1 = TKNOT - 1; w = 0.0f;
        S = Kb[TKNOT - 1] + (double)(t - 47903) * (double)f0b[TKNOT - 1];
    } else {
        int s = (t - 96) / 192;
        int r = t - (192 * s + 96);      // 0..191 within segment
        i0 = s; i1 = s + 1;
        w = ((float)r + 0.5f) * (1.0f / 192.0f);
        double fs  = (double)f0b[s];
        double fs1 = (double)f0b[s + 1];
        double rp  = (double)(r + 1);
        S = Kb[s] + rp * fs + (fs1 - fs) * rp * rp * (1.0 / 384.0);
    }
    const double PhiT = S * (1.0 / (double)NSAMP);   // phase in turns
    const float  f0up = f0b[i0] * (1.0f - w) + f0b[i1] * w;

    const float* row0 = ha + (size_t)b * TKNOT * NHARM + (size_t)i0 * NHARM;
    const float* row1 = ha + (size_t)b * TKNOT * NHARM + (size_t)i1 * NHARM;

    v8f acc = {};                        // 16x16 f32 accumulator (8 VGPRs)
    v2f aone; aone.x = 1.0f; aone.y = 1.0f;   // all-ones 16x4 A matrix

    // 25 chunks of 4 harmonics; half 0 -> local K {0,1}, half 1 -> {2,3}
    for (int kb = 0; kb < NHARM; kb += 4) {
        v2f bm;
        #pragma unroll
        for (int v = 0; v < 2; ++v) {
            const int kidx = kb + 2 * half + v;       // 0-based harmonic index
            const double cd = Cb[kidx];               // k*sqrt(1+inh*k^2)
            const float  cf = (float)cd;
            const float  mask = (f0up * cf < 21600.0f) ? 1.0f : 0.0f;
            const double u    = PhiT * cd;            // phase in turns
            const double frac = u - floor(u);         // [0, 1)
            const float  sv   = __builtin_amdgcn_sinf((float)frac); // v_sin_f32
            const float  amp  = row0[kidx] * (1.0f - w) + row1[kidx] * w;
            const float  p    = amp * mask * sv;
            if (v == 0) bm.x = p; else bm.y = p;
        }
        // D = A(ones) x B + C  ->  every row of C accumulates sum over 4 K
        acc = __builtin_amdgcn_wmma_f32_16x16x4_f32(
            false, aone, false, bm, (short)0, acc, false, false);
    }

    // C VGPR0: lanes 0-15 hold (M=0, N=lane); lanes 16-31 duplicate (M=8).
    if (half == 0) {
        out[(size_t)b * NSAMP + t] = acc[0] * (1.0f / (float)NHARM);
    }
}

// ---------------------------------------------------------------------------
extern "C" void kernel_launch(void* const* d_in, const int* in_sizes, int n_in,
                              void* d_out, int out_size, void* d_ws, size_t ws_size,
                              hipStream_t stream) {
    const float* harm_amps = (const float*)d_in[0];  // (16, 250, 100) f32
    const float* f0_hz     = (const float*)d_in[1];  // (16, 250) f32
    const float* inh       = (const float*)d_in[2];  // (16,) f32
    // d_in[3] = n_samples (int, == 48000) -- shapes are fixed constants here.
    float*  out = (float*)d_out;                     // (16, 1, 48000) f32
    double* K   = (double*)d_ws;                     // 16*250*8 = 32 KB
    double* C   = K + BATCH * TKNOT;                 // 16*100*8 = 12.8 KB

    hsynth_setup_kernel<<<1, 256, 0, stream>>>(f0_hz, inh, K, C);

    // 16 batches * 3000 tiles = 48000 waves; 8 waves/block -> 6000 blocks
    hsynth_synth_kernel<<<(BATCH * (NSAMP / 16)) / 8, 256, 0, stream>>>(
        harm_amps, f0_hz, K, C, out);
}